// MSE_loss_64871186039089
// MI455X (gfx1250) — compile-verified
//
#include <hip/hip_runtime.h>
#include <hip/hip_bf16.h>

typedef __attribute__((ext_vector_type(2))) float v2f;
typedef __attribute__((ext_vector_type(8))) float v8f;

#define HW   311296   // 256*1216 elements per (batch, channel)
#define HW4  77824    // HW/4
#define NBIN 9        // weight_map values 0..8

// ---------------------------------------------------------------------------
// Stage 1: streaming binned reduction.
// Per-wave per-lane LDS histogram: pair index = wave*288 + bin*32 + lane.
// dword index = 2*(wave*288 + bin*32 + lane) -> bank = 2*lane mod 64, so each
// lane owns banks {2l,2l+1}: ds_load_b64/ds_store_b64 are conflict-free for
// any bin pattern. Output: per-workgroup partials, 32 floats per WG:
//   cols 0..8  = bin sums, cols 16..24 = bin counts, rest zero.
// ---------------------------------------------------------------------------
__global__ __launch_bounds__(256) void mse_hist_stage1(
    const float* __restrict__ pred, const float* __restrict__ gt,
    const int* __restrict__ wm, float* __restrict__ partial, int n4)
{
  __shared__ float2 hist[8 * NBIN * 32];
  const int t        = threadIdx.x;
  const int lane     = t & 31;
  const int waveBase = (t >> 5) * (NBIN * 32);

  for (int i = t; i < 8 * NBIN * 32; i += 256) hist[i] = make_float2(0.f, 0.f);
  __syncthreads();

  const float4* __restrict__ p4 = (const float4*)pred;
  const float4* __restrict__ g4 = (const float4*)gt;
  const int4*   __restrict__ w4 = (const int4*)wm;

  const int gstride = gridDim.x * 256;
  for (int i4 = blockIdx.x * 256 + t; i4 < n4; i4 += gstride) {
    const int b = i4 / HW4;                 // batch index (const-div -> mulhi)
    const float4 pv = p4[i4 + b * HW4];     // prediction[:,0] vec4
    const float4 gv = g4[i4];
    const int4   wv = w4[i4];

    // speculative prefetch of next grid-stride iteration (global_prefetch_b8)
    __builtin_prefetch(&g4[i4 + gstride], 0, 1);
    __builtin_prefetch(&w4[i4 + gstride], 0, 1);
    __builtin_prefetch(&p4[i4 + gstride + b * HW4], 0, 1);

    {
      float e = pv.x - gv.x; int idx = waveBase + wv.x * 32 + lane;
      float2 v = hist[idx]; v.x += e * e; v.y += 1.f; hist[idx] = v;
    }
    {
      float e = pv.y - gv.y; int idx = waveBase + wv.y * 32 + lane;
      float2 v = hist[idx]; v.x += e * e; v.y += 1.f; hist[idx] = v;
    }
    {
      float e = pv.z - gv.z; int idx = waveBase + wv.z * 32 + lane;
      float2 v = hist[idx]; v.x += e * e; v.y += 1.f; hist[idx] = v;
    }
    {
      float e = pv.w - gv.w; int idx = waveBase + wv.w * 32 + lane;
      float2 v = hist[idx]; v.x += e * e; v.y += 1.f; hist[idx] = v;
    }
  }
  __syncthreads();

  // fold 8 wave regions into wave-0 region (each thread owns disjoint slots)
  for (int p = t; p < NBIN * 32; p += 256) {
    float2 a = hist[p];
    #pragma unroll
    for (int w = 1; w < 8; ++w) {
      float2 bb = hist[w * NBIN * 32 + p];
      a.x += bb.x; a.y += bb.y;
    }
    hist[p] = a;
  }
  __syncthreads();

  if (t < 16) {
    float S = 0.f, C = 0.f;
    if (t < NBIN) {
      #pragma unroll
      for (int l = 0; l < 32; ++l) {
        float2 v = hist[t * 32 + l];
        S += v.x; C += v.y;
      }
    }
    partial[blockIdx.x * 32 + t]      = S;  // cols 0..15 (9..15 zero)
    partial[blockIdx.x * 32 + 16 + t] = C;  // cols 16..31 (25..31 zero)
  }
}

// ---------------------------------------------------------------------------
// Stage 2: deterministic final reduction. Fold num_wg rows -> 8 rows in LDS,
// then one wave sums those rows on the matrix unit:
//   D = ones(16x4) x B(4x16) + C  =>  D[m,n] = sum_k B[k,n]   (all m equal)
// A == ones makes the result invariant to the exact K-lane striping, so this
// is robust. Column n of D = total for partial-matrix column n.
// ---------------------------------------------------------------------------
__global__ __launch_bounds__(256) void mse_hist_stage2(
    const float* __restrict__ partial, float* __restrict__ out, int num_wg)
{
  __shared__ float red[8 * 32];
  const int t   = threadIdx.x;
  const int col = t & 31;
  const int g   = t >> 5;

  float acc = 0.f;
  for (int r = g; r < num_wg; r += 8) acc += partial[r * 32 + col];
  red[g * 32 + col] = acc;
  __syncthreads();

  if (t < 32) {   // wave 0 only; uniform-true -> EXEC all ones for WMMA
    v8f cS = {0.f,0.f,0.f,0.f,0.f,0.f,0.f,0.f};
    v8f cC = {0.f,0.f,0.f,0.f,0.f,0.f,0.f,0.f};
    const v2f A = {1.f, 1.f};
    const int cc = t & 15, half = t >> 4;
    #pragma unroll
    for (int r0 = 0; r0 < 8; r0 += 4) {
      const int row = r0 + 2 * half;
      v2f bS = { red[row * 32 + cc],      red[(row + 1) * 32 + cc] };
      v2f bC = { red[row * 32 + 16 + cc], red[(row + 1) * 32 + 16 + cc] };
      cS = __builtin_amdgcn_wmma_f32_16x16x4_f32(false, A, false, bS,
                                                 (short)0, cS, false, false);
      cC = __builtin_amdgcn_wmma_f32_16x16x4_f32(false, A, false, bC,
                                                 (short)0, cC, false, false);
    }
    // lanes 1..8 hold (S_k, C_k) for bins 1..8 in element 0 of the D tile
    float m = (t >= 1 && t <= 8) ? (cS[0] / fmaxf(cC[0], 1.0f)) : 0.0f;
    #pragma unroll
    for (int off = 16; off >= 1; off >>= 1) m += __shfl_xor(m, off, 32);
    if (t == 0) out[0] = m * 0.125f;   // sum(means[1:9]) / MAX_VAL
  }
}

extern "C" void kernel_launch(void* const* d_in, const int* in_sizes, int n_in,
                              void* d_out, int out_size, void* d_ws, size_t ws_size,
                              hipStream_t stream) {
  const float* pred = (const float*)d_in[0];   // [32,2,256,1216] f32
  const float* gt   = (const float*)d_in[1];   // [32,1,256,1216] f32
  const int*   wm   = (const int*)d_in[2];     // [32,1,256,1216] i32
  float*       out  = (float*)d_out;

  const int N  = in_sizes[1];   // B*H*W = 9,961,472 (divisible by 4)
  const int n4 = N / 4;

  int num_wg = 2048;
  const size_t row_bytes = 32 * sizeof(float);
  if (ws_size < (size_t)num_wg * row_bytes) {
    num_wg = (int)(ws_size / row_bytes);
    if (num_wg < 1) num_wg = 1;
  }
  float* partial = (float*)d_ws;

  mse_hist_stage1<<<num_wg, 256, 0, stream>>>(pred, gt, wm, partial, n4);
  mse_hist_stage2<<<1, 256, 0, stream>>>(partial, out, num_wg);
}